// HebbianLayer_35622458753122
// MI455X (gfx1250) — compile-verified
//
#include <hip/hip_runtime.h>

#define IN_F  512
#define OUT_F 128
#define BATCH 4096
#define LRATE 1e-4f

typedef __attribute__((ext_vector_type(2))) float v2f;
typedef __attribute__((ext_vector_type(4))) float v4f;
typedef __attribute__((ext_vector_type(8))) float v8f;

// ---------------------------------------------------------------------------
// Kernel 1: y = x @ W0^T + bias via V_WMMA_F32_16X16X4_F32 (fp32 WMMA).
// Grid: BATCH/16 blocks of 256 threads (8 wave32s). Wave w computes the
// 16x16 output tile at rows [16*blockIdx.x, +16), cols [16*w, +16).
// A-matrix layout (16x4 f32): lane L -> M = L%16, VGPR v -> K = 2*(L/16)+v.
// B-matrix (4x16, B[k][n] = W[n][k]): lane L -> N = L%16, VGPR v -> K = 2*(L/16)+v.
// ---------------------------------------------------------------------------
__global__ __launch_bounds__(256) void hebb_fwd_gemm(
    const float* __restrict__ x, const float* __restrict__ w,
    const float* __restrict__ bias, float* __restrict__ y) {
  const int lane = threadIdx.x & 31;
  const int wid  = threadIdx.x >> 5;
  const int g    = lane >> 4;      // half-wave selector
  const int l16  = lane & 15;
  const int m0   = blockIdx.x * 16;
  const int n0   = wid * 16;

  const float* arow = x + (size_t)(m0 + l16) * IN_F + 2 * g;
  const float* brow = w + (size_t)(n0 + l16) * IN_F + 2 * g;  // B[k][n] = W[n][k]

  v8f acc = {};
  #pragma unroll 8
  for (int kb = 0; kb < IN_F; kb += 4) {
    v2f a = *(const v2f*)(arow + kb);
    v2f b = *(const v2f*)(brow + kb);
    // 8 args: (neg_a, A, neg_b, B, c_mod, C, reuse_a, reuse_b)
    acc = __builtin_amdgcn_wmma_f32_16x16x4_f32(false, a, false, b,
                                                (short)0, acc, false, false);
  }

  const float bv = bias[n0 + l16];
  #pragma unroll
  for (int r = 0; r < 8; ++r) {
    y[(size_t)(m0 + r + 8 * g) * OUT_F + n0 + l16] = acc[r] + bv;
  }
}

// ---------------------------------------------------------------------------
// Kernel 2: sequential Oja/Hebbian scan. Single workgroup, 512 threads
// (16 wave32s). The full 128x512 fp32 weight matrix (256 KB) lives in the
// 320 KB WGP LDS for the entire 4096-step scan (zero HBM traffic for state).
// Per sample:
//   phase 1 (all 16 waves): ys[i] = dot(W[i,:], xs) + bias[i]
//       - b128 row reads (lane l covers cols 128j+4l..+3, bank-conflict-free)
//       - shfl_xor wave reduction
//   phase 2 (128 threads, 4 columns each via v4f):
//       p_i = sum_{j<=i} ys[j]*W[j,c]   (inclusive prefix over rows)
//       W[i,c] += LR * ys[i] * (xs[c] - p_i)
//   In-place update is exact: row i enters the prefix before being
//   overwritten; rows j>i are read before their own update.
// ---------------------------------------------------------------------------
__global__ __launch_bounds__(IN_F) void hebb_scan(
    const float* __restrict__ x, const float* __restrict__ w0,
    const float* __restrict__ bias, float* __restrict__ wout) {
  extern __shared__ float lds[];
  float* Wl  = lds;                         // OUT_F * IN_F
  float* xsh = Wl  + OUT_F * IN_F;          // IN_F
  float* ysh = xsh + IN_F;                  // OUT_F
  float* bsh = ysh + OUT_F;                 // OUT_F

  const int t    = threadIdx.x;             // 0..511
  const int lane = t & 31;
  const int wid  = t >> 5;

  // Stage initial weights (b128) + bias into LDS.
  {
    v4f* Wv = (v4f*)Wl;
    const v4f* w0v = (const v4f*)w0;
    for (int i = t; i < OUT_F * IN_F / 4; i += IN_F) Wv[i] = w0v[i];
  }
  if (t < OUT_F) bsh[t] = bias[t];
  __syncthreads();

  float xv = x[t];  // sample 0, column t

  for (int b = 0; b < BATCH; ++b) {
    xsh[t] = xv;
    __syncthreads();  // xs visible; previous step's W updates visible

    // Cache this lane's xs slice once (reused across 8 dot products).
    v4f xr[4];
    #pragma unroll
    for (int j = 0; j < 4; ++j) xr[j] = *(const v4f*)(xsh + 128 * j + 4 * lane);

    // ys: wave wid computes outputs [8*wid, 8*wid+8)
    #pragma unroll
    for (int r = 0; r < 8; ++r) {
      const int i = wid * 8 + r;
      const float* wrow = Wl + i * IN_F;
      float acc = 0.f;
      #pragma unroll
      for (int j = 0; j < 4; ++j) {
        v4f wq = *(const v4f*)(wrow + 128 * j + 4 * lane);
        acc = fmaf(wq[0], xr[j][0], acc);
        acc = fmaf(wq[1], xr[j][1], acc);
        acc = fmaf(wq[2], xr[j][2], acc);
        acc = fmaf(wq[3], xr[j][3], acc);
      }
      acc += __shfl_xor(acc, 16, 32);
      acc += __shfl_xor(acc, 8, 32);
      acc += __shfl_xor(acc, 4, 32);
      acc += __shfl_xor(acc, 2, 32);
      acc += __shfl_xor(acc, 1, 32);
      if (lane == 0) ysh[i] = acc + bsh[i];
    }
    __syncthreads();  // ys visible

    // Prefetch next sample while doing the update pass.
    const float xn = (b + 1 < BATCH) ? x[(size_t)(b + 1) * IN_F + t] : 0.f;

    // Fused prefix + update: 128 threads, thread t owns columns 4t..4t+3.
    if (t < OUT_F) {
      const int c0 = 4 * t;
      const v4f xq = *(const v4f*)(xsh + c0);
      v4f p = {0.f, 0.f, 0.f, 0.f};
      #pragma unroll 2
      for (int i = 0; i < OUT_F; i += 4) {
        const v4f y4 = *(const v4f*)(ysh + i);
        #pragma unroll
        for (int u = 0; u < 4; ++u) {
          const float yi = y4[u];
          float* wp = Wl + (i + u) * IN_F + c0;
          v4f wq = *(v4f*)wp;
          p[0] = fmaf(yi, wq[0], p[0]);   // inclusive prefix (tril incl. diag)
          p[1] = fmaf(yi, wq[1], p[1]);
          p[2] = fmaf(yi, wq[2], p[2]);
          p[3] = fmaf(yi, wq[3], p[3]);
          const float s = LRATE * yi;
          wq[0] = fmaf(s, xq[0] - p[0], wq[0]);
          wq[1] = fmaf(s, xq[1] - p[1], wq[1]);
          wq[2] = fmaf(s, xq[2] - p[2], wq[2]);
          wq[3] = fmaf(s, xq[3] - p[3], wq[3]);
          *(v4f*)wp = wq;
        }
      }
    }
    xv = xn;
    // Next iteration's xsh write + barrier orders everything.
  }

  __syncthreads();
  {
    const v4f* Wv = (const v4f*)Wl;
    v4f* wo = (v4f*)wout;
    for (int i = t; i < OUT_F * IN_F / 4; i += IN_F) wo[i] = Wv[i];
  }
}

// ---------------------------------------------------------------------------
extern "C" void kernel_launch(void* const* d_in, const int* in_sizes, int n_in,
                              void* d_out, int out_size, void* d_ws, size_t ws_size,
                              hipStream_t stream) {
  (void)in_sizes; (void)n_in; (void)out_size; (void)d_ws; (void)ws_size;
  const float* x    = (const float*)d_in[0];   // [BATCH, IN_F]
  const float* w    = (const float*)d_in[1];   // [OUT_F, IN_F]
  const float* bias = (const float*)d_in[2];   // [OUT_F]
  float* y    = (float*)d_out;                 // [BATCH, OUT_F]
  float* wout = y + (size_t)BATCH * OUT_F;     // [OUT_F, IN_F]

  hebb_fwd_gemm<<<BATCH / 16, 256, 0, stream>>>(x, w, bias, y);

  const size_t shbytes = (size_t)(OUT_F * IN_F + IN_F + 2 * OUT_F) * sizeof(float);
  hebb_scan<<<1, IN_F, shbytes, stream>>>(x, w, bias, wout);
}